// SparseAttentionEncoder_56143812494060
// MI455X (gfx1250) — compile-verified
//
#include <hip/hip_runtime.h>
#include <hip/hip_bf16.h>
#include <math.h>

typedef __attribute__((ext_vector_type(2))) float v2f;
typedef __attribute__((ext_vector_type(4))) float v4f;
typedef __attribute__((ext_vector_type(8))) float v8f;

#define HWC 2304   // H*W = 48*48, fixed by the reference
#define EC  256

__device__ __forceinline__ float gelu_exact(float v) {
    return 0.5f * v * (1.0f + erff(v * 0.70710678118654752f));
}

// ---------------------------------------------------------------------------
// WMMA fp32 GEMM over pixels:  Y[n,m] = sum_k W[n,k]*X[k,m] (+bias)(+gelu)(+res)
// One wave computes a 64x16 (m,n) block: 4 m-tiles sharing the B (weight)
// fragment per K-step -> 4 v_wmma_f32_16x16x4_f32 per 9 loads.
// M is compile-time (HWC) so all strides are immediate; pointers bump by
// constant byte offsets (no per-iteration 64-bit multiplies).
// grid = (M/64, N/16, B)
// ---------------------------------------------------------------------------
__global__ void gemm16_wmma(const float* __restrict__ Wt,   // (N,K) row-major
                            const float* __restrict__ X,    // (B,K,M)
                            const float* __restrict__ bias, // (N) or null
                            float* __restrict__ Y,          // (B,N,M) via sYb
                            const float* __restrict__ res,  // (B,N,M) or null
                            int N, int K, long sXb, long sYb, int act)
{
    constexpr int M = HWC;
    const int lane = threadIdx.x & 31;
    const int hi   = lane >> 4;
    const int l    = lane & 15;
    const int m0   = blockIdx.x << 6;      // 64 rows per wave
    const int n0   = blockIdx.y << 4;
    const int b    = blockIdx.z;

    X += (long)b * sXb;
    Y += (long)b * sYb;
    if (res) res += (long)b * sYb;

    const int ka = 2 * hi;
    const float* xp = X + (long)ka * M + m0 + l;          // A: rows ka, ka+1
    const float* wp = Wt + (long)(n0 + l) * K + ka;       // B: col n0+l

    v8f acc0 = {}, acc1 = {}, acc2 = {}, acc3 = {};
    for (int k0 = 0; k0 < K; k0 += 4) {
        const v2f bf = *(const v2f*)wp;
        v2f a0, a1, a2, a3;
        a0.x = xp[0];      a0.y = xp[M];
        a1.x = xp[16];     a1.y = xp[M + 16];
        a2.x = xp[32];     a2.y = xp[M + 32];
        a3.x = xp[48];     a3.y = xp[M + 48];
        __builtin_prefetch(xp + 8 * M, 0, 1);
        acc0 = __builtin_amdgcn_wmma_f32_16x16x4_f32(false, a0, false, bf, (short)0, acc0, false, false);
        acc1 = __builtin_amdgcn_wmma_f32_16x16x4_f32(false, a1, false, bf, (short)0, acc1, false, false);
        acc2 = __builtin_amdgcn_wmma_f32_16x16x4_f32(false, a2, false, bf, (short)0, acc2, false, false);
        acc3 = __builtin_amdgcn_wmma_f32_16x16x4_f32(false, a3, false, bf, (short)0, acc3, false, false);
        xp += 4 * M;
        wp += 4;
    }

    const float bn = bias ? bias[n0 + l] : 0.0f;
    float*       yc = Y + (long)(n0 + l) * M + m0 + 8 * hi;
    const float* rc = res ? res + (long)(n0 + l) * M + m0 + 8 * hi : nullptr;
    v8f accs[4] = {acc0, acc1, acc2, acc3};
#pragma unroll
    for (int t = 0; t < 4; ++t) {
        v4f lo, hi4;
#pragma unroll
        for (int r = 0; r < 4; ++r) {
            float v = accs[t][r] + bn;
            if (act == 1) v = gelu_exact(v);
            lo[r] = v;
            float w = accs[t][r + 4] + bn;
            if (act == 1) w = gelu_exact(w);
            hi4[r] = w;
        }
        if (rc) {
            const v4f r0 = *(const v4f*)(rc + 16 * t);
            const v4f r1 = *(const v4f*)(rc + 16 * t + 4);
            lo += r0; hi4 += r1;
        }
        *(v4f*)(yc + 16 * t)     = lo;
        *(v4f*)(yc + 16 * t + 4) = hi4;
    }
}

// ---------------------------------------------------------------------------
// Attention: one wave per (16-query tile, head, batch). HW/E compile-time.
// Phase 1: S = (Q^T K)*scale + bias -> LDS strip (16 x HW, rows padded +4)
//          Q fragments hoisted into 8 registers (invariant over j-tiles).
// Phase 2: exact softmax, wave shfl reductions (single wave, no barriers).
// Phase 3: O = P * V^T, two accumulators (d halves), b64 operand loads.
// ---------------------------------------------------------------------------
__global__ void attn_wmma(const float* __restrict__ qkv, // (B, 3E, HW)
                          const float* __restrict__ obs, // (B, HW) 0/1
                          float* __restrict__ out)       // (B, E, HW)
{
    constexpr int HW = HWC;
    constexpr int E  = EC;
    constexpr int RS = HW + 4;           // LDS row stride: kill 64-bank conflicts
    extern __shared__ float smem[];      // 16 * RS floats

    const int lane = threadIdx.x & 31;
    const int hi   = lane >> 4;
    const int l    = lane & 15;
    const int i0   = blockIdx.x << 4;
    const int h    = blockIdx.y;
    const int b    = blockIdx.z;
    const float scale = 0.17677669529663688f; // 1/sqrt(32)

    const float* Q  = qkv + ((long)b * 3 * E + h * 32) * HW;
    const float* Kp = Q + (long)E * HW;
    const float* V  = Q + (long)2 * E * HW;
    const float* ob = obs + (long)b * HW;

    float obsi[8];
#pragma unroll
    for (int r = 0; r < 8; ++r) obsi[r] = ob[i0 + r + 8 * hi];

    // Hoist Q fragments: qf[kk] covers K-dim [4kk, 4kk+4)
    v2f qf[8];
    {
        const float* qp = Q + (long)(2 * hi) * HW + i0 + l;
#pragma unroll
        for (int kk = 0; kk < 8; ++kk) {
            qf[kk].x = qp[(long)(4 * kk) * HW];
            qf[kk].y = qp[(long)(4 * kk + 1) * HW];
        }
    }

    // ---- Phase 1: scores ----
    const float* kpb = Kp + (long)(2 * hi) * HW + l;
    for (int j0 = 0; j0 < HW; j0 += 16) {
        v8f s = {};
#pragma unroll
        for (int kk = 0; kk < 8; ++kk) {
            v2f bf;
            bf.x = kpb[(long)(4 * kk) * HW + j0];
            bf.y = kpb[(long)(4 * kk + 1) * HW + j0];
            s = __builtin_amdgcn_wmma_f32_16x16x4_f32(
                    false, qf[kk], false, bf, (short)0, s, false, false);
        }
        const float obsj = ob[j0 + l];
#pragma unroll
        for (int r = 0; r < 8; ++r) {
            const float bv = (obsi[r] * obsj == 0.0f) ? -10000.0f : 0.0f;
            smem[(r + 8 * hi) * RS + j0 + l] = s[r] * scale + bv;
        }
    }

    // ---- Phase 2: softmax (single wave: DS counters keep order) ----
    for (int r = 0; r < 16; ++r) {
        float* row = smem + r * RS;
        float mx = -3.402823466e38f;
        for (int j = lane; j < HW; j += 32) mx = fmaxf(mx, row[j]);
#pragma unroll
        for (int o = 16; o > 0; o >>= 1) mx = fmaxf(mx, __shfl_xor(mx, o, 32));
        float sum = 0.0f;
        for (int j = lane; j < HW; j += 32) {
            const float e = expf(row[j] - mx);
            row[j] = e;
            sum += e;
        }
#pragma unroll
        for (int o = 16; o > 0; o >>= 1) sum += __shfl_xor(sum, o, 32);
        const float inv = 1.0f / sum;
        for (int j = lane; j < HW; j += 32) row[j] *= inv;
    }

    // ---- Phase 3: O = P * V^T ----
    v8f o0 = {}, o1 = {};
    {
        const float* ap = smem + l * RS + 2 * hi;          // P row l
        const float* v0 = V + (long)l * HW + 2 * hi;       // d = l
        const float* v1 = V + (long)(16 + l) * HW + 2 * hi;// d = 16+l
        for (int j0 = 0; j0 < HW; j0 += 4) {
            const v2f a  = *(const v2f*)(ap + j0);
            const v2f b0 = *(const v2f*)(v0 + j0);
            const v2f b1 = *(const v2f*)(v1 + j0);
            o0 = __builtin_amdgcn_wmma_f32_16x16x4_f32(
                     false, a, false, b0, (short)0, o0, false, false);
            o1 = __builtin_amdgcn_wmma_f32_16x16x4_f32(
                     false, a, false, b1, (short)0, o1, false, false);
        }
    }
    float* outb = out + ((long)b * E + h * 32) * HW + i0 + 8 * hi;
    v4f s00, s01, s10, s11;
#pragma unroll
    for (int r = 0; r < 4; ++r) {
        s00[r] = o0[r]; s01[r] = o0[r + 4];
        s10[r] = o1[r]; s11[r] = o1[r + 4];
    }
    *(v4f*)(outb + (long)l * HW)            = s00;
    *(v4f*)(outb + (long)l * HW + 4)        = s01;
    *(v4f*)(outb + (long)(16 + l) * HW)     = s10;
    *(v4f*)(outb + (long)(16 + l) * HW + 4) = s11;
}

// ---------------------------------------------------------------------------
// GroupNorm (optionally fused exact GELU). grid = (groups, B), block = 256.
// ---------------------------------------------------------------------------
__global__ void gn_kernel(const float* __restrict__ X, float* __restrict__ Y,
                          const float* __restrict__ gamma,
                          const float* __restrict__ beta,
                          int C, int HW, int groups, long sB, int doGelu)
{
    __shared__ float red[256];
    const int g = blockIdx.x, b = blockIdx.y, tid = threadIdx.x;
    const int cpg = C / groups;
    const long N = (long)cpg * HW;
    const float* xb = X + (long)b * sB + (long)g * cpg * HW;
    float*       yb = Y + (long)b * sB + (long)g * cpg * HW;

    float s = 0.0f, s2 = 0.0f;
    for (long i = tid; i < N; i += 256) {
        const float v = xb[i];
        s += v; s2 += v * v;
    }
    red[tid] = s; __syncthreads();
    for (int o = 128; o > 0; o >>= 1) { if (tid < o) red[tid] += red[tid + o]; __syncthreads(); }
    const float mu = red[0] / (float)N; __syncthreads();
    red[tid] = s2; __syncthreads();
    for (int o = 128; o > 0; o >>= 1) { if (tid < o) red[tid] += red[tid + o]; __syncthreads(); }
    const float var = red[0] / (float)N - mu * mu;
    const float inv = rsqrtf(var + 1e-5f);
    __syncthreads();

    for (long i = tid; i < N; i += 256) {
        const int c = g * cpg + (int)(i / HW);
        float v = (xb[i] - mu) * inv * gamma[c] + beta[c];
        if (doGelu) v = gelu_exact(v);
        yb[i] = v;
    }
}

// First conv of the tiny embed branches (C_in = 1 or 2).
__global__ void small_c1(const float* __restrict__ X, long sXb, int Cin,
                         const float* __restrict__ Wt, const float* __restrict__ bias,
                         float* __restrict__ Y, int Cout, int HW, int B)
{
    const long total = (long)B * Cout * HW;
    for (long idx = blockIdx.x * (long)blockDim.x + threadIdx.x; idx < total;
         idx += (long)gridDim.x * blockDim.x) {
        const int b = (int)(idx / ((long)Cout * HW));
        const long rem = idx - (long)b * Cout * HW;
        const int c = (int)(rem / HW);
        const int m = (int)(rem % HW);
        float acc = bias[c];
        for (int ci = 0; ci < Cin; ++ci)
            acc += Wt[c * Cin + ci] * X[(long)b * sXb + (long)ci * HW + m];
        Y[idx] = acc;
    }
}

// 7x7 box dilation of (mask > 0.5) -> 0/1 float.
__global__ void dilate_kernel(const float* __restrict__ mask, float* __restrict__ obs,
                              int B, int H, int W)
{
    const int HW = H * W;
    const long total = (long)B * HW;
    const long idx = blockIdx.x * (long)blockDim.x + threadIdx.x;
    if (idx >= total) return;
    const int b = (int)(idx / HW);
    const int p = (int)(idx % HW);
    const int y = p / W, x = p % W;
    float r = 0.0f;
    for (int dy = -3; dy <= 3; ++dy) {
        const int yy = y + dy; if (yy < 0 || yy >= H) continue;
        for (int dx = -3; dx <= 3; ++dx) {
            const int xx = x + dx; if (xx < 0 || xx >= W) continue;
            if (mask[(long)b * HW + yy * W + xx] > 0.5f) r = 1.0f;
        }
    }
    obs[idx] = r;
}

// ---------------------------------------------------------------------------
extern "C" void kernel_launch(void* const* d_in, const int* in_sizes, int n_in,
                              void* d_out, int out_size, void* d_ws, size_t ws_size,
                              hipStream_t stream) {
    const int B = 2, H = 48, Wd = 48, HW = H * Wd;   // HW == HWC
    const int E = 256, SD = 128, CD = 64, MD = 32;
    const int CAT = E + SD + CD + MD; // 480

    const float* x      = (const float*)d_in[0];
    const float* coords = (const float*)d_in[1];
    const float* mask   = (const float*)d_in[2];
    const float* in_w   = (const float*)d_in[3];
    const float* in_b   = (const float*)d_in[4];
    const float* sen_w1 = (const float*)d_in[5];
    const float* sen_b1 = (const float*)d_in[6];
    const float* sen_gw = (const float*)d_in[7];
    const float* sen_gb = (const float*)d_in[8];
    const float* sen_w2 = (const float*)d_in[9];
    const float* sen_b2 = (const float*)d_in[10];
    const float* co_w1  = (const float*)d_in[11];
    const float* co_b1  = (const float*)d_in[12];
    const float* co_gw  = (const float*)d_in[13];
    const float* co_gb  = (const float*)d_in[14];
    const float* co_w2  = (const float*)d_in[15];
    const float* co_b2  = (const float*)d_in[16];
    const float* mk_w1  = (const float*)d_in[17];
    const float* mk_b1  = (const float*)d_in[18];
    const float* mk_gw  = (const float*)d_in[19];
    const float* mk_gb  = (const float*)d_in[20];
    const float* mk_w2  = (const float*)d_in[21];
    const float* mk_b2  = (const float*)d_in[22];
    const float* fus_w  = (const float*)d_in[23];
    const float* fus_b  = (const float*)d_in[24];
    const float* fus_gw = (const float*)d_in[25];
    const float* fus_gb = (const float*)d_in[26];
    const float* qkv_w  = (const float*)d_in[27];
    const float* qkv_b  = (const float*)d_in[28];
    const float* out_w  = (const float*)d_in[29];
    const float* out_b  = (const float*)d_in[30];
    const float* ffn_w1 = (const float*)d_in[31];
    const float* ffn_b1 = (const float*)d_in[32];
    const float* ffn_w2 = (const float*)d_in[33];
    const float* ffn_b2 = (const float*)d_in[34];
    const float* n1_w   = (const float*)d_in[35];
    const float* n1_b   = (const float*)d_in[36];
    const float* n2_w   = (const float*)d_in[37];
    const float* n2_b   = (const float*)d_in[38];
    const float* op_w   = (const float*)d_in[39];
    const float* op_b   = (const float*)d_in[40];

    float* ws = (float*)d_ws;
    size_t o_xcat = 0;
    size_t o_emb  = o_xcat + (size_t)B * CAT * HW;
    size_t o_fus  = o_emb  + (size_t)B * SD * HW;
    size_t o_n    = o_fus  + (size_t)B * E * HW;
    size_t o_qkv  = o_n    + (size_t)B * E * HW;
    size_t o_attn = o_qkv  + (size_t)B * 3 * E * HW;
    size_t o_x1   = o_attn + (size_t)B * E * HW;
    size_t o_ffn  = o_x1   + (size_t)B * E * HW;
    size_t o_x2   = o_ffn  + (size_t)B * 4 * E * HW;
    size_t o_obs  = o_x2   + (size_t)B * E * HW;
    (void)ws_size; (void)n_in; (void)in_sizes; (void)out_size;

    const dim3 w32(32);
    const int MB = HW / 64; // 36 m-blocks (64 rows each)

    // --- sen embed: baseline = x[:, :1] ---
    small_c1<<<(B * SD * HW + 255) / 256, 256, 0, stream>>>(
        x, (long)4 * HW, 1, sen_w1, sen_b1, ws + o_emb, SD, HW, B);
    gn_kernel<<<dim3(8, B), 256, 0, stream>>>(
        ws + o_emb, ws + o_emb, sen_gw, sen_gb, SD, HW, 8, (long)SD * HW, 1);
    gemm16_wmma<<<dim3(MB, SD / 16, B), w32, 0, stream>>>(
        sen_w2, ws + o_emb, sen_b2, ws + o_xcat + (size_t)E * HW, nullptr,
        SD, SD, (long)SD * HW, (long)CAT * HW, 0);

    // --- co embed: coords (2 ch) ---
    small_c1<<<(B * CD * HW + 255) / 256, 256, 0, stream>>>(
        coords, (long)2 * HW, 2, co_w1, co_b1, ws + o_emb, CD, HW, B);
    gn_kernel<<<dim3(8, B), 256, 0, stream>>>(
        ws + o_emb, ws + o_emb, co_gw, co_gb, CD, HW, 8, (long)CD * HW, 1);
    gemm16_wmma<<<dim3(MB, CD / 16, B), w32, 0, stream>>>(
        co_w2, ws + o_emb, co_b2, ws + o_xcat + (size_t)(E + SD) * HW, nullptr,
        CD, CD, (long)CD * HW, (long)CAT * HW, 0);

    // --- mk embed: mask (1 ch) ---
    small_c1<<<(B * MD * HW + 255) / 256, 256, 0, stream>>>(
        mask, (long)HW, 1, mk_w1, mk_b1, ws + o_emb, MD, HW, B);
    gn_kernel<<<dim3(8, B), 256, 0, stream>>>(
        ws + o_emb, ws + o_emb, mk_gw, mk_gb, MD, HW, 8, (long)MD * HW, 1);
    gemm16_wmma<<<dim3(MB, MD / 16, B), w32, 0, stream>>>(
        mk_w2, ws + o_emb, mk_b2, ws + o_xcat + (size_t)(E + SD + CD) * HW, nullptr,
        MD, MD, (long)MD * HW, (long)CAT * HW, 0);

    // --- x_proj (K=4) into xcat[:, 0:256] ---
    gemm16_wmma<<<dim3(MB, E / 16, B), w32, 0, stream>>>(
        in_w, x, in_b, ws + o_xcat, nullptr,
        E, 4, (long)4 * HW, (long)CAT * HW, 0);

    // --- fusion conv (K=480) + GN + GELU -> residual ---
    gemm16_wmma<<<dim3(MB, E / 16, B), w32, 0, stream>>>(
        fus_w, ws + o_xcat, fus_b, ws + o_fus, nullptr,
        E, CAT, (long)CAT * HW, (long)E * HW, 0);
    gn_kernel<<<dim3(8, B), 256, 0, stream>>>(
        ws + o_fus, ws + o_fus, fus_gw, fus_gb, E, HW, 8, (long)E * HW, 1);

    // --- n1 = GN(fused), qkv projection ---
    gn_kernel<<<dim3(8, B), 256, 0, stream>>>(
        ws + o_fus, ws + o_n, n1_w, n1_b, E, HW, 8, (long)E * HW, 0);
    gemm16_wmma<<<dim3(MB, (3 * E) / 16, B), w32, 0, stream>>>(
        qkv_w, ws + o_n, qkv_b, ws + o_qkv, nullptr,
        3 * E, E, (long)E * HW, (long)3 * E * HW, 0);

    // --- mask dilation + attention ---
    dilate_kernel<<<(B * HW + 255) / 256, 256, 0, stream>>>(
        mask, ws + o_obs, B, H, Wd);
    const size_t smem_bytes = (size_t)16 * (HW + 4) * sizeof(float);
    attn_wmma<<<dim3(HW / 16, 8, B), w32, smem_bytes, stream>>>(
        ws + o_qkv, ws + o_obs, ws + o_attn);

    // --- out projection + residual -> x1 ---
    gemm16_wmma<<<dim3(MB, E / 16, B), w32, 0, stream>>>(
        out_w, ws + o_attn, out_b, ws + o_x1, ws + o_fus,
        E, E, (long)E * HW, (long)E * HW, 0);

    // --- n2 = GN(x1), FFN ---
    gn_kernel<<<dim3(8, B), 256, 0, stream>>>(
        ws + o_x1, ws + o_n, n2_w, n2_b, E, HW, 8, (long)E * HW, 0);
    gemm16_wmma<<<dim3(MB, (4 * E) / 16, B), w32, 0, stream>>>(
        ffn_w1, ws + o_n, ffn_b1, ws + o_ffn, nullptr,
        4 * E, E, (long)E * HW, (long)4 * E * HW, 1);
    gemm16_wmma<<<dim3(MB, E / 16, B), w32, 0, stream>>>(
        ffn_w2, ws + o_ffn, ffn_b2, ws + o_x2, ws + o_x1,
        E, 4 * E, (long)4 * E * HW, (long)E * HW, 0);

    // --- final projection -> d_out ---
    gemm16_wmma<<<dim3(MB, E / 16, B), w32, 0, stream>>>(
        op_w, ws + o_x2, op_b, (float*)d_out, nullptr,
        E, E, (long)E * HW, (long)E * HW, 0);
}